// KLoRALinearLayer_37400575213757
// MI455X (gfx1250) — compile-verified
//
#include <hip/hip_runtime.h>
#include <hip/hip_bf16.h>
#include <math.h>

typedef __attribute__((ext_vector_type(2))) float v2f;
typedef __attribute__((ext_vector_type(4))) float v4f;
typedef __attribute__((ext_vector_type(8))) float v8f;

#define D           4096
#define LRANK       8
#define TILE_M      16
#define KTOP        64
#define NB_TOPK     128
#define TOPK_THREADS 128

// ---------------------------------------------------------------------------
// Kernel 1: per-block partial top-64 of |a @ b| for both candidate matrices.
// blockIdx.y selects matrix (0 -> w1, 1 -> w2). Each thread fixes a column i
// (so the 8-entry b-column lives in registers) and walks rows o; it maintains
// a replace-min top-64 list in its private LDS segment. The block then
// extracts its top-64 by 64 rounds of argmax reduction.
// ---------------------------------------------------------------------------
__global__ __launch_bounds__(TOPK_THREADS)
void topk_partial_kernel(const float* __restrict__ w1a, const float* __restrict__ w1b,
                         const float* __restrict__ w2a, const float* __restrict__ w2b,
                         float* __restrict__ cand)
{
    __shared__ float seg[TOPK_THREADS * KTOP];   // 32 KB
    __shared__ float red[TOPK_THREADS];
    __shared__ int   redIdx[TOPK_THREADS];

    const int tid = threadIdx.x;
    const int mat = blockIdx.y;
    const float* __restrict__ A  = mat ? w2a : w1a;
    const float* __restrict__ Bw = mat ? w2b : w1b;

    #pragma unroll
    for (int k = 0; k < KTOP; ++k) seg[tid * KTOP + k] = 0.0f;
    float curMin = 0.0f;
    int   minIdx = 0;

    const int gtid = blockIdx.x * TOPK_THREADS + tid;
    const int i    = gtid & (D - 1);

    float bcol[LRANK];
    #pragma unroll
    for (int r = 0; r < LRANK; ++r) bcol[r] = Bw[r * D + i];

    const int totalThreads = NB_TOPK * TOPK_THREADS;     // 16384
    const int iters        = (D * D) / totalThreads;     // 1024
    const int oStep        = totalThreads / D;           // 4
    int o = gtid >> 12;

    for (int it = 0; it < iters; ++it, o += oStep) {
        float val = 0.0f;
        #pragma unroll
        for (int r = 0; r < LRANK; ++r) val += A[o * LRANK + r] * bcol[r];
        val = fabsf(val);
        if (val > curMin) {
            seg[tid * KTOP + minIdx] = val;
            curMin = 3.4e38f;
            #pragma unroll
            for (int k = 0; k < KTOP; ++k) {
                float v = seg[tid * KTOP + k];
                if (v < curMin) { curMin = v; minIdx = k; }
            }
        }
    }
    __syncthreads();

    float* outp = cand + mat * (NB_TOPK * KTOP) + blockIdx.x * KTOP;
    for (int round = 0; round < KTOP; ++round) {
        float lm = -1.0f; int li = 0;
        for (int k = 0; k < KTOP; ++k) {
            float v = seg[tid * KTOP + k];
            if (v > lm) { lm = v; li = k; }
        }
        red[tid] = lm; redIdx[tid] = tid * KTOP + li;
        __syncthreads();
        for (int s = TOPK_THREADS / 2; s > 0; s >>= 1) {
            if (tid < s && red[tid + s] > red[tid]) {
                red[tid] = red[tid + s]; redIdx[tid] = redIdx[tid + s];
            }
            __syncthreads();
        }
        if (tid == 0) { outp[round] = red[0]; seg[redIdx[0]] = -1.0f; }
        __syncthreads();
    }
}

// ---------------------------------------------------------------------------
// Kernel 2: merge 128x64 candidates per matrix, sum global top-64, apply the
// timestep scale and write the selection flag.
// ---------------------------------------------------------------------------
__global__ __launch_bounds__(TOPK_THREADS)
void topk_final_kernel(const float* __restrict__ cand,
                       const int* __restrict__ tsPtr,
                       int* __restrict__ flag)
{
    __shared__ float seg[NB_TOPK * KTOP];        // 8192 floats = 32 KB
    __shared__ float red[TOPK_THREADS];
    __shared__ int   redIdx[TOPK_THREADS];
    __shared__ float sums[2];

    const int tid = threadIdx.x;
    const int per = (NB_TOPK * KTOP) / TOPK_THREADS;   // 64

    for (int m = 0; m < 2; ++m) {
        for (int k = 0; k < per; ++k)
            seg[tid * per + k] = cand[m * NB_TOPK * KTOP + tid * per + k];
        __syncthreads();
        float acc = 0.0f;
        for (int round = 0; round < KTOP; ++round) {
            float lm = -1.0f; int li = 0;
            for (int k = 0; k < per; ++k) {
                float v = seg[tid * per + k];
                if (v > lm) { lm = v; li = k; }
            }
            red[tid] = lm; redIdx[tid] = tid * per + li;
            __syncthreads();
            for (int s = TOPK_THREADS / 2; s > 0; s >>= 1) {
                if (tid < s && red[tid + s] > red[tid]) {
                    red[tid] = red[tid + s]; redIdx[tid] = redIdx[tid + s];
                }
                __syncthreads();
            }
            if (tid == 0) { acc += red[0]; seg[redIdx[0]] = -1.0f; }
            __syncthreads();
        }
        if (tid == 0) sums[m] = acc;
        __syncthreads();
    }
    if (tid == 0) {
        int t = tsPtr[0] % 28000;
        float scale = fmodf(1.5f * (float)t / 28000.0f + 0.5f, 1.5f);
        float temp_ratio = sums[0] / (sums[1] * scale);
        *flag = (temp_ratio > 1.0f) ? 1 : 0;
    }
}

// ---------------------------------------------------------------------------
// Kernel 3: fused low-rank GEMM.
//   Stage 1: tmp[16,8] = h_tile[16,4096] @ b^T  (NT float4 loads, shfl reduce)
//   Stage 2: out_tile[16,4096] = tmp @ a^T via v_wmma_f32_16x16x4_f32,
//            NT stores (single-use streaming output; keep a/b RT in L2).
// ---------------------------------------------------------------------------
__global__ __launch_bounds__(256)
void lora_gemm_kernel(const float* __restrict__ h,
                      const float* __restrict__ w1a, const float* __restrict__ w1b,
                      const float* __restrict__ w2a, const float* __restrict__ w2b,
                      const int* __restrict__ flag,
                      float* __restrict__ out)
{
    __shared__ float tmpS[TILE_M][LRANK];

    const int tid  = threadIdx.x;
    const int lane = tid & 31;
    const int wave = tid >> 5;                   // 0..7
    const int rowBase = blockIdx.x * TILE_M;

    const int sel = *flag;
    const float* __restrict__ aW = sel ? w1a : w2a;
    const float* __restrict__ bW = sel ? w1b : w2b;

    // ---------------- Stage 1: tmp[m][r] = sum_i h[row,i] * b[r,i] ----------
    {
        const int m = tid >> 4;                  // 0..15, lanes 0-15 / 16-31 split
        const int j = tid & 15;                  // 0..15 column group
        const float* hrow = h + (size_t)(rowBase + m) * D;
        float acc[LRANK];
        #pragma unroll
        for (int r = 0; r < LRANK; ++r) acc[r] = 0.0f;

        for (int it = 0; it < D / 64; ++it) {    // 64 iterations of float4
            int i0 = (j + 16 * it) * 4;
            // h is streamed exactly once -> non-temporal (don't rinse L2)
            v4f h4 = __builtin_nontemporal_load((const v4f*)(hrow + i0));
            #pragma unroll
            for (int r = 0; r < LRANK; ++r) {
                v4f b4 = *(const v4f*)(bW + r * D + i0);   // reused -> RT cached
                acc[r] += h4.x * b4.x + h4.y * b4.y + h4.z * b4.z + h4.w * b4.w;
            }
        }
        // reduce across the 16 lanes sharing a row (stays within half-wave)
        #pragma unroll
        for (int r = 0; r < LRANK; ++r) {
            float v = acc[r];
            v += __shfl_xor(v, 1, 32);
            v += __shfl_xor(v, 2, 32);
            v += __shfl_xor(v, 4, 32);
            v += __shfl_xor(v, 8, 32);
            acc[r] = v;
        }
        if (j == 0) {
            #pragma unroll
            for (int r = 0; r < LRANK; ++r) tmpS[m][r] = acc[r];
        }
    }
    __syncthreads();

    // ---------------- Stage 2: out = tmp @ a^T via WMMA f32 16x16x4 ----------
    // A (16x4 f32) layout: lane%16 = M, VGPR j holds K = j + 2*(lane>>4)
    // B (4x16 f32) layout: lane%16 = N, VGPR j holds K = j + 2*(lane>>4)
    // C (16x16 f32): VGPR j holds M = j + 8*(lane>>4), N = lane%16
    const int n  = lane & 15;
    const int kh = lane >> 4;                    // 0 or 1

    v2f a0, a1;
    a0.x = tmpS[n][2 * kh];     a0.y = tmpS[n][2 * kh + 1];
    a1.x = tmpS[n][4 + 2 * kh]; a1.y = tmpS[n][5 + 2 * kh];

    for (int t = 0; t < (D / TILE_M) / 8; ++t) { // 32 o-tiles per wave
        const int ob = (wave + 8 * t) * TILE_M;
        const int o  = ob + n;
        v2f b0 = *(const v2f*)(aW + o * LRANK + 2 * kh);      // K = 0..3
        v2f b1 = *(const v2f*)(aW + o * LRANK + 4 + 2 * kh);  // K = 4..7
        v8f c = {};
        c = __builtin_amdgcn_wmma_f32_16x16x4_f32(false, a0, false, b0,
                                                  (short)0, c, false, false);
        c = __builtin_amdgcn_wmma_f32_16x16x4_f32(false, a1, false, b1,
                                                  (short)0, c, false, false);
        float* obase = out + (size_t)rowBase * D + ob;
        #pragma unroll
        for (int jj = 0; jj < 8; ++jj) {
            // out is written exactly once -> non-temporal store
            __builtin_nontemporal_store(c[jj], &obase[(size_t)(jj + 8 * kh) * D + n]);
        }
    }
}

// ---------------------------------------------------------------------------
extern "C" void kernel_launch(void* const* d_in, const int* in_sizes, int n_in,
                              void* d_out, int out_size, void* d_ws, size_t ws_size,
                              hipStream_t stream)
{
    const float* h   = (const float*)d_in[0];
    const float* w1a = (const float*)d_in[1];
    const float* w1b = (const float*)d_in[2];
    const float* w2a = (const float*)d_in[3];
    const float* w2b = (const float*)d_in[4];
    const int*   ts  = (const int*)d_in[5];
    float* out = (float*)d_out;

    float* cand = (float*)d_ws;                                   // 2*128*64 floats
    int*   flag = (int*)((char*)d_ws + 2 * NB_TOPK * KTOP * sizeof(float));

    const int rows = in_sizes[0] / D;                             // B*S = 8192

    dim3 g1(NB_TOPK, 2);
    topk_partial_kernel<<<g1, TOPK_THREADS, 0, stream>>>(w1a, w1b, w2a, w2b, cand);
    topk_final_kernel<<<1, TOPK_THREADS, 0, stream>>>(cand, ts, flag);
    lora_gemm_kernel<<<rows / TILE_M, 256, 0, stream>>>(h, w1a, w1b, w2a, w2b, flag, out);
}